// SelfAttention_91319594647720
// MI455X (gfx1250) — compile-verified
//
#include <hip/hip_runtime.h>
#include <stdint.h>

// Self-attention with full attn-matrix output, fp32 exact path.
// B=64, S=1024, D=64.  scores = QK^T/8, mask -> -inf, softmax, out = P @ V.
// V_WMMA_F32_16X16X4_F32 for both GEMMs; softmax in unshifted exp2 domain
// (safe: |scaled score| << 127), per-lane partial sums merged once.

#define BB 64
#define SS 1024
#define DD 64
// (1/TEMPERATURE) * log2(e): softmax computed as exp2 of scaled scores
#define K2SCALE (0.125f * 1.44269504088896340736f)

typedef float v2f __attribute__((ext_vector_type(2)));
typedef float v8f __attribute__((ext_vector_type(8)));

static __device__ __forceinline__ v8f wmma_f32x4(v2f a, v2f b, v8f c) {
    // (neg_a, A, neg_b, B, c_mod, C, reuse_a, reuse_b)
    return __builtin_amdgcn_wmma_f32_16x16x4_f32(false, a, false, b,
                                                 (short)0, c, false, false);
}

// raw v_exp_f32 (exp2);  exp2(-inf) = 0 handles masked entries for free
static __device__ __forceinline__ float exp2_fast(float x) {
    return __builtin_amdgcn_exp2f(x);
}

__global__ __launch_bounds__(128) void
attn_fused_kernel(const float* __restrict__ q,
                  const float* __restrict__ k,
                  const float* __restrict__ v,
                  const uint8_t* __restrict__ mask,
                  float* __restrict__ out,
                  float* __restrict__ attn) {
    // 4 waves/block; each wave owns one 16-query tile.
    __shared__ float pbuf[4][16 * 18];   // padded (stride 18) P-tile per wave

    const int tid  = threadIdx.x;
    const int wid  = tid >> 5;
    const int lane = tid & 31;
    const int hi   = (lane >> 4) & 1;    // which half of the wave
    const int n16  = lane & 15;

    const int b     = blockIdx.x >> 4;         // batch
    const int qblk  = blockIdx.x & 15;         // 64-query block within batch
    const int qwave = qblk * 64 + wid * 16;    // first query row of this wave

    // ---- Load Q tile as WMMA-A fragments: a[j] covers K-dims 4j..4j+3 ----
    // A layout (16x4 f32): lane m=lane%16 holds {K=2hi, K=2hi+1} -> float2.
    const float* qrow = q + ((size_t)(b * SS + qwave + n16)) * DD + 2 * hi;
    v2f a[16];
#pragma unroll
    for (int j = 0; j < 16; ++j)
        a[j] = *(const v2f*)(qrow + 4 * j);

    // mask/attn element for (row r, kv-tile t): mrow[r*S + t*16]
    const uint8_t* mrow = mask + ((size_t)(b * SS + qwave + 8 * hi)) * SS + n16;

    // per-LANE partial softmax denominators (exp2 domain, unshifted);
    // lane owns columns kv = t*16 + n16, t = 0..63, for its 8 rows.
    float l8[8];
#pragma unroll
    for (int r = 0; r < 8; ++r) l8[r] = 0.0f;

    // ================= PASS 1: per-lane denominator sums =================
    for (int t = 0; t < SS / 16; ++t) {
        // two interleaved accumulators break the WMMA RAW chain
        v8f c0 = {}, c1 = {};
        const float* krow = k + ((size_t)(b * SS + t * 16 + n16)) * DD + 2 * hi;
#pragma unroll
        for (int j = 0; j < 16; j += 2) {
            v2f b0 = *(const v2f*)(krow + 4 * j);
            v2f b1 = *(const v2f*)(krow + 4 * j + 4);
            c0 = wmma_f32x4(a[j],     b0, c0);
            c1 = wmma_f32x4(a[j + 1], b1, c1);
        }
        v8f c = c0 + c1;
#pragma unroll
        for (int r = 0; r < 8; ++r) {
            float s = c[r] * K2SCALE;
            if (mrow[(size_t)r * SS + (size_t)t * 16]) s = -INFINITY;
            l8[r] += exp2_fast(s);                  // masked -> +0
        }
    }

    // ---- one-time cross-lane sum over the 16-lane half ----
#pragma unroll
    for (int r = 0; r < 8; ++r) {
        float l = l8[r];
#pragma unroll
        for (int d = 1; d < 16; d <<= 1)
            l += __shfl_xor(l, d, 32);
        l8[r] = l;
    }

    // normalization; fully-masked rows (l==0) -> p = 0 everywhere
    float invl[8];
#pragma unroll
    for (int r = 0; r < 8; ++r)
        invl[r] = (l8[r] > 0.0f) ? 1.0f / l8[r] : 0.0f;

    // =================== PASS 2: write attn, out += P@V ===================
    v8f o[4];
#pragma unroll
    for (int nt = 0; nt < 4; ++nt) o[nt] = (v8f){};

    float* pb   = &pbuf[wid][0];
    float* arow = attn + ((size_t)(b * SS + qwave + 8 * hi)) * SS + n16;

    for (int t = 0; t < SS / 16; ++t) {
        v8f c0 = {}, c1 = {};
        const float* krow = k + ((size_t)(b * SS + t * 16 + n16)) * DD + 2 * hi;
#pragma unroll
        for (int j = 0; j < 16; j += 2) {            // identical to pass 1
            v2f b0 = *(const v2f*)(krow + 4 * j);
            v2f b1 = *(const v2f*)(krow + 4 * j + 4);
            c0 = wmma_f32x4(a[j],     b0, c0);
            c1 = wmma_f32x4(a[j + 1], b1, c1);
        }
        v8f c = c0 + c1;
#pragma unroll
        for (int r = 0; r < 8; ++r) {
            float s = c[r] * K2SCALE;
            if (mrow[(size_t)r * SS + (size_t)t * 16]) s = -INFINITY;
            float p = exp2_fast(s) * invl[r];        // 0 for masked/dead rows
            arow[(size_t)r * SS + (size_t)t * 16] = p;
            pb[(r + 8 * hi) * 18 + n16] = p;         // C-layout -> LDS
        }
        __syncthreads();                             // LDS write -> read order
        // P tile (A layout) x V tile (B layout); o[0..3] interleave 4 chains
#pragma unroll
        for (int j = 0; j < 4; ++j) {
            v2f pa = *(const v2f*)(pb + n16 * 18 + 4 * j + 2 * hi);
            const float* vrow =
                v + ((size_t)(b * SS + t * 16 + 4 * j + 2 * hi)) * DD + n16;
#pragma unroll
            for (int nt = 0; nt < 4; ++nt) {
                v2f vb;
                vb.x = vrow[nt * 16];                // V[kv  ][nt*16+n16]
                vb.y = vrow[DD + nt * 16];           // V[kv+1][nt*16+n16]
                o[nt] = wmma_f32x4(pa, vb, o[nt]);
            }
        }
        __syncthreads();                             // LDS reuse next iter
    }

    // ---- store output tile [16 x 64] ----
    const size_t obase = ((size_t)(b * SS + qwave + 8 * hi)) * DD + n16;
#pragma unroll
    for (int nt = 0; nt < 4; ++nt)
#pragma unroll
        for (int r = 0; r < 8; ++r)
            out[obase + (size_t)r * DD + nt * 16] = o[nt][r];
}

extern "C" void kernel_launch(void* const* d_in, const int* in_sizes, int n_in,
                              void* d_out, int out_size, void* d_ws, size_t ws_size,
                              hipStream_t stream) {
    const float*   q    = (const float*)d_in[0];
    const float*   k    = (const float*)d_in[1];
    const float*   v    = (const float*)d_in[2];
    const uint8_t* mask = (const uint8_t*)d_in[3];   // jnp.bool_ -> 1 byte/elem

    float* out  = (float*)d_out;                     // [B,S,D]
    float* attn = out + (size_t)BB * SS * DD;        // [B,S,S]

    dim3 grid(BB * (SS / 64));   // 64 queries per block (4 waves x 16)
    dim3 block(128);
    attn_fused_kernel<<<grid, block, 0, stream>>>(q, k, v, mask, out, attn);
}